// G_43662637531680
// MI455X (gfx1250) — compile-verified
//
#include <hip/hip_runtime.h>
#include <math.h>

typedef float v2f __attribute__((ext_vector_type(2)));
typedef float v8f __attribute__((ext_vector_type(8)));

#define B_   128
#define T_   64
#define E_   512
#define H_   512
#define V_   10000
#define FIN_ 2048

__device__ __forceinline__ v2f ld2(const float* p) { return *(const v2f*)p; }

__device__ __forceinline__ v8f wmma4(v2f a, v2f b, v8f c) {
  // V_WMMA_F32_16X16X4_F32 : D = A(16x4) * B(4x16) + C(16x16), fp32
  return __builtin_amdgcn_wmma_f32_16x16x4_f32(false, a, false, b, (short)0, c,
                                               false, false);
}

// ---------------------------------------------------------------------------
// Generic C[M,N] = cat(A1[M,K1], A2[M,K2]) @ W[N,K1+K2]^T + bias
// One wave32 per (16 x 16*NB) output strip: the A fragment is loaded once per
// K-step and reused across NB WMMAs (register blocking along N) to raise the
// WMMA : VMEM issue ratio.
// Fragment layout (ISA 7.12.2, 32-bit):
//   A 16x4: lanes 0-15 -> row=lane, K = k,k+1 ; lanes 16-31 -> K = k+2,k+3
//   B 4x16: lanes 0-15 -> col=lane, K = k,k+1 ; lanes 16-31 -> K = k+2,k+3
//   C/D   : VGPR r -> row = r (lanes 0-15) or r+8 (lanes 16-31), col = lane%16
// ---------------------------------------------------------------------------
template <int NB>
__global__ void concat_gemm_wmma(const float* __restrict__ A1,
                                 const float* __restrict__ A2,
                                 const float* __restrict__ W,
                                 const float* __restrict__ bias,
                                 float* __restrict__ C,
                                 int M, int N, int K1, int K2,
                                 int nGroupsN, int totalGroups)
{
  int wave  = threadIdx.x >> 5;
  int group = blockIdx.x * (blockDim.x >> 5) + wave;
  if (group >= totalGroups) return;            // wave-uniform: EXEC stays full
  int m0   = (group / nGroupsN) * 16;
  int n0   = (group % nGroupsN) * (16 * NB);
  int lane = threadIdx.x & 31;
  int half = lane >> 4;
  int l    = lane & 15;
  int koff = half * 2;
  int K    = K1 + K2;

  const float* wrow[NB];
#pragma unroll
  for (int i = 0; i < NB; ++i)
    wrow[i] = W + (size_t)(n0 + i * 16 + l) * K;
  const float* a1row = A1 + (size_t)(m0 + l) * K1;

  // warm L0 for the streaming operands (global_prefetch_b8)
  __builtin_prefetch(a1row, 0, 3);
#pragma unroll
  for (int i = 0; i < NB; ++i) __builtin_prefetch(wrow[i], 0, 3);

  v8f acc[NB];
#pragma unroll
  for (int i = 0; i < NB; ++i) acc[i] = (v8f){};

  for (int k = 0; k < K1; k += 4) {
    v2f a = ld2(a1row + k + koff);
#pragma unroll
    for (int i = 0; i < NB; ++i)
      acc[i] = wmma4(a, ld2(wrow[i] + k + koff), acc[i]);
  }
  if (K2 > 0) {
    const float* a2row = A2 + (size_t)(m0 + l) * K2;
    __builtin_prefetch(a2row, 0, 3);
    for (int k = 0; k < K2; k += 4) {
      v2f a = ld2(a2row + k + koff);
#pragma unroll
      for (int i = 0; i < NB; ++i)
        acc[i] = wmma4(a, ld2(wrow[i] + K1 + k + koff), acc[i]);
    }
  }

#pragma unroll
  for (int i = 0; i < NB; ++i) {
    int col = n0 + i * 16 + l;
    float bb = bias ? bias[col] : 0.0f;
#pragma unroll
    for (int r = 0; r < 8; ++r) {
      int row = m0 + r + half * 8;
      C[(size_t)row * N + col] = acc[i][r] + bb;
    }
  }
}

// ---------------------------------------------------------------------------
// BatchNorm over batch axis: per-column mean / rsqrt(var+eps) of f[B_,E_]
// ---------------------------------------------------------------------------
__global__ void bn_stats_kernel(const float* __restrict__ f,
                                float* __restrict__ mu,
                                float* __restrict__ rstd)
{
  __shared__ float ss[128];
  __shared__ float sq[128];
  int e = blockIdx.x;
  int b = threadIdx.x;                 // 128 threads = B_
  float v = f[(size_t)b * E_ + e];
  ss[b] = v;
  sq[b] = v * v;
  __syncthreads();
  for (int s = 64; s > 0; s >>= 1) {
    if (b < s) { ss[b] += ss[b + s]; sq[b] += sq[b + s]; }
    __syncthreads();
  }
  if (b == 0) {
    float m   = ss[0] * (1.0f / 128.0f);
    float var = sq[0] * (1.0f / 128.0f) - m * m;
    mu[e]   = m;
    rstd[e] = rsqrtf(var + 1e-5f);
  }
}

__global__ void bn_apply_kernel(const float* __restrict__ f,
                                const float* __restrict__ mu,
                                const float* __restrict__ rstd,
                                const float* __restrict__ gamma,
                                const float* __restrict__ beta,
                                float* __restrict__ feats)
{
  int i = blockIdx.x * blockDim.x + threadIdx.x;
  if (i >= B_ * E_) return;
  int e = i % E_;
  feats[i] = gamma[e] * (f[i] - mu[e]) * rstd[e] + beta[e];
}

// x_seq[t][b][:] = embed_W[captions[b][t-1]] for t = 1..T-1
__global__ void embed_kernel(const int* __restrict__ captions,
                             const float* __restrict__ embW,
                             float* __restrict__ xseq)
{
  long i = (long)blockIdx.x * blockDim.x + threadIdx.x;
  const long total = (long)(T_ - 1) * B_ * E_;
  if (i >= total) return;
  int  e   = (int)(i % E_);
  long be  = i / E_;
  int  b   = (int)(be % B_);
  int  tm1 = (int)(be / B_);           // 0..T-2  ->  t = tm1+1
  int  tok = captions[b * T_ + tm1];
  xseq[(long)(tm1 + 1) * B_ * E_ + (long)b * E_ + e] =
      embW[(long)tok * E_ + e];
}

// row-softmax of attn logits, then applied = feats * softmax(row)
__global__ void softmax_applied_kernel(const float* __restrict__ alog,
                                       const float* __restrict__ feats,
                                       float* __restrict__ applied)
{
  __shared__ float red[256];
  int b   = blockIdx.x;
  int tid = threadIdx.x;               // 256 threads
  const float* row = alog + (size_t)b * H_;

  float m = -1e30f;
  for (int j = tid; j < H_; j += 256) m = fmaxf(m, row[j]);
  red[tid] = m; __syncthreads();
  for (int s = 128; s > 0; s >>= 1) {
    if (tid < s) red[tid] = fmaxf(red[tid], red[tid + s]);
    __syncthreads();
  }
  m = red[0]; __syncthreads();

  float sum = 0.0f;
  for (int j = tid; j < H_; j += 256) sum += expf(row[j] - m);
  red[tid] = sum; __syncthreads();
  for (int s = 128; s > 0; s >>= 1) {
    if (tid < s) red[tid] += red[tid + s];
    __syncthreads();
  }
  float inv = 1.0f / red[0];

  for (int j = tid; j < H_; j += 256) {
    float aw = expf(row[j] - m) * inv;
    applied[(size_t)b * E_ + j] = feats[(size_t)b * E_ + j] * aw;
  }
}

// ---------------------------------------------------------------------------
// Fused GRU step: 6 WMMA accumulators (xr,xz,xn over Wih; hr,hz,hn over Whh)
// sharing two A fragments per K-step, then gate math; writes hs[t]
// (time-major, for packed logits) and hiddens[b, t, :] (batch-major output).
// ---------------------------------------------------------------------------
__global__ void gru_step_wmma(const float* __restrict__ inp,
                              const float* __restrict__ hprev,
                              const float* __restrict__ Wih,
                              const float* __restrict__ Whh,
                              const float* __restrict__ bih,
                              const float* __restrict__ bhh,
                              float* __restrict__ hs_t,
                              float* __restrict__ hiddens,
                              int t)
{
  int wave = threadIdx.x >> 5;
  int tile = blockIdx.x * (blockDim.x >> 5) + wave;   // 256 tiles total, exact
  int m0   = (tile / (H_ / 16)) * 16;
  int n0   = (tile % (H_ / 16)) * 16;
  int lane = threadIdx.x & 31;
  int half = lane >> 4;
  int l    = lane & 15;
  int koff = half * 2;

  const float* wr = Wih + (size_t)(n0 + l) * H_;
  const float* wz = Wih + (size_t)(H_ + n0 + l) * H_;
  const float* wn = Wih + (size_t)(2 * H_ + n0 + l) * H_;
  const float* ur = Whh + (size_t)(n0 + l) * H_;
  const float* uz = Whh + (size_t)(H_ + n0 + l) * H_;
  const float* un = Whh + (size_t)(2 * H_ + n0 + l) * H_;
  const float* arow = inp   + (size_t)(m0 + l) * H_;
  const float* hrow = hprev + (size_t)(m0 + l) * H_;

  __builtin_prefetch(arow, 0, 3);
  __builtin_prefetch(hrow, 0, 3);

  v8f axr = {}, axz = {}, axn = {}, ahr = {}, ahz = {}, ahn = {};
  for (int k = 0; k < H_; k += 4) {
    v2f a = ld2(arow + k + koff);
    axr = wmma4(a, ld2(wr + k + koff), axr);
    axz = wmma4(a, ld2(wz + k + koff), axz);
    axn = wmma4(a, ld2(wn + k + koff), axn);
    v2f h = ld2(hrow + k + koff);
    ahr = wmma4(h, ld2(ur + k + koff), ahr);
    ahz = wmma4(h, ld2(uz + k + koff), ahz);
    ahn = wmma4(h, ld2(un + k + koff), ahn);
  }

  int j = n0 + l;
  float br = bih[j], bz = bih[H_ + j], bn = bih[2 * H_ + j];
  float cr = bhh[j], cz = bhh[H_ + j], cn = bhh[2 * H_ + j];
#pragma unroll
  for (int r = 0; r < 8; ++r) {
    int row = m0 + r + half * 8;
    float xr = axr[r] + br, hr = ahr[r] + cr;
    float xz = axz[r] + bz, hz = ahz[r] + cz;
    float xn = axn[r] + bn, hn = ahn[r] + cn;
    float rg = 1.0f / (1.0f + expf(-(xr + hr)));
    float zg = 1.0f / (1.0f + expf(-(xz + hz)));
    float nt = tanhf(xn + rg * hn);
    float hv = (1.0f - zg) * nt + zg * hprev[(size_t)row * H_ + j];
    hs_t[(size_t)row * H_ + j] = hv;
    hiddens[(size_t)row * T_ * H_ + (size_t)t * H_ + j] = hv;
  }
}

// in-place row log-softmax over V_ columns
__global__ void log_softmax_kernel(float* __restrict__ logits)
{
  __shared__ float red[256];
  float* row = logits + (size_t)blockIdx.x * V_;
  int tid = threadIdx.x;               // 256

  float m = -1e30f;
  for (int j = tid; j < V_; j += 256) m = fmaxf(m, row[j]);
  red[tid] = m; __syncthreads();
  for (int s = 128; s > 0; s >>= 1) {
    if (tid < s) red[tid] = fmaxf(red[tid], red[tid + s]);
    __syncthreads();
  }
  m = red[0]; __syncthreads();

  float sum = 0.0f;
  for (int j = tid; j < V_; j += 256) sum += expf(row[j] - m);
  red[tid] = sum; __syncthreads();
  for (int s = 128; s > 0; s >>= 1) {
    if (tid < s) red[tid] += red[tid + s];
    __syncthreads();
  }
  float lse = m + logf(red[0]);

  for (int j = tid; j < V_; j += 256) row[j] = row[j] - lse;
}

// ---------------------------------------------------------------------------
extern "C" void kernel_launch(void* const* d_in, const int* in_sizes, int n_in,
                              void* d_out, int out_size, void* d_ws, size_t ws_size,
                              hipStream_t stream)
{
  const float* features = (const float*)d_in[0];
  const int*   captions = (const int*)  d_in[1];
  const float* h0       = (const float*)d_in[2];
  // d_in[3] = lengths (== T_), compile-time constant here
  const float* embed_W  = (const float*)d_in[4];
  const float* fc_W     = (const float*)d_in[5];
  const float* fc_b     = (const float*)d_in[6];
  const float* bn_gamma = (const float*)d_in[7];
  const float* bn_beta  = (const float*)d_in[8];
  const float* attn_W   = (const float*)d_in[9];
  const float* attn_b   = (const float*)d_in[10];
  const float* comb_W   = (const float*)d_in[11];
  const float* comb_b   = (const float*)d_in[12];
  const float* gru_Wih  = (const float*)d_in[13];
  const float* gru_Whh  = (const float*)d_in[14];
  const float* gru_bih  = (const float*)d_in[15];
  const float* gru_bhh  = (const float*)d_in[16];
  const float* out_W    = (const float*)d_in[17];
  const float* out_b    = (const float*)d_in[18];

  float* ws      = (float*)d_ws;
  float* xseq    = ws;                              // [T,B,E] (slot 0 = feats)
  float* hs      = xseq + (size_t)T_ * B_ * E_;     // [T,B,H] time-major
  float* fbuf    = hs   + (size_t)T_ * B_ * H_;     // [B,E] pre-BN fc output
  float* mu      = fbuf + (size_t)B_ * E_;          // [E]
  float* rstd    = mu + E_;                         // [E]
  float* awbuf   = rstd + E_;                       // [B,H] attn logits
  float* applied = awbuf   + (size_t)B_ * H_;       // [B,E]
  float* inpbuf  = applied + (size_t)B_ * E_;       // [B,H]

  float* logits  = (float*)d_out;                   // [T*B, V] -> log-probs
  float* hiddens = logits + (size_t)T_ * B_ * V_;   // [B, T, H]

  // 1) fc: f = features @ fc_W^T + fc_b   (M=128, N=512, K=2048), NB=4
  {
    int nG = (E_ / 16) / 4, tot = (B_ / 16) * nG;   // 8 * 8 = 64 groups
    concat_gemm_wmma<4><<<(tot + 3) / 4, 128, 0, stream>>>(
        features, features, fc_W, fc_b, fbuf, B_, E_, FIN_, 0, nG, tot);
  }
  // 2) BatchNorm (training-mode batch stats) -> feats = xseq slot 0
  bn_stats_kernel<<<E_, 128, 0, stream>>>(fbuf, mu, rstd);
  bn_apply_kernel<<<(B_ * E_ + 255) / 256, 256, 0, stream>>>(
      fbuf, mu, rstd, bn_gamma, bn_beta, xseq);
  // 3) embedding gather into xseq[1..T-1]
  {
    long total = (long)(T_ - 1) * B_ * E_;
    embed_kernel<<<(int)((total + 255) / 256), 256, 0, stream>>>(
        captions, embed_W, xseq);
  }
  // 4) recurrence
  for (int t = 0; t < T_; ++t) {
    const float* xt = xseq + (size_t)t * B_ * E_;
    const float* hp = (t == 0) ? h0 : hs + (size_t)(t - 1) * B_ * H_;
    int nG = (H_ / 16) / 4, tot = (B_ / 16) * nG;   // 64 groups -> 16 blocks
    // attn logits = cat(x, h) @ attn_W^T + attn_b
    concat_gemm_wmma<4><<<tot / 4, 128, 0, stream>>>(
        xt, hp, attn_W, attn_b, awbuf, B_, H_, E_, H_, nG, tot);
    // softmax + applied = feats * aw
    softmax_applied_kernel<<<B_, 256, 0, stream>>>(awbuf, xseq, applied);
    // inp = cat(x, applied) @ comb_W^T + comb_b
    concat_gemm_wmma<4><<<tot / 4, 128, 0, stream>>>(
        xt, applied, comb_W, comb_b, inpbuf, B_, H_, E_, E_, nG, tot);
    // fused GRU GEMMs + gates (256 tiles -> 64 blocks)
    gru_step_wmma<<<64, 128, 0, stream>>>(
        inpbuf, hp, gru_Wih, gru_Whh, gru_bih, gru_bhh,
        hs + (size_t)t * B_ * H_, hiddens, t);
  }
  // 5) vocab projection: logits = hs_flat @ out_W^T + out_b
  //    M=8192, N=10000 -> 625 N-tiles = 125 groups of NB=5 (exact)
  {
    int nG = (V_ / 16) / 5;                         // 125
    int tot = (T_ * B_ / 16) * nG;                  // 512 * 125 = 64000 groups
    concat_gemm_wmma<5><<<(tot + 3) / 4, 128, 0, stream>>>(
        hs, hs, out_W, out_b, logits, T_ * B_, V_, H_, 0, nG, tot);
  }
  // 6) in-place log-softmax over V
  log_softmax_kernel<<<T_ * B_, 256, 0, stream>>>(logits);
}